// GCNWithSelfAttention_25082609008775
// MI455X (gfx1250) — compile-verified
//
#include <hip/hip_runtime.h>

typedef float v2f __attribute__((ext_vector_type(2)));
typedef float v8f __attribute__((ext_vector_type(8)));
typedef int   v4i __attribute__((ext_vector_type(4)));

static constexpr int Bsz = 4, Nn = 2048, Din = 256, Hh = 512, DoutC = 256;
static constexpr int Ee = 65536, NH = 8, DH = 64;

__device__ __forceinline__ v8f wmma4(v2f a, v2f b, v8f c) {
  // D = A(16x4,f32) * B(4x16,f32) + C(16x16,f32)
  return __builtin_amdgcn_wmma_f32_16x16x4_f32(
      /*neg_a=*/false, a, /*neg_b=*/false, b,
      /*c_mod=*/(short)0, c, /*reuse_a=*/false, /*reuse_b=*/false);
}

__device__ __forceinline__ v8f v8zero() {
  v8f r;
#pragma unroll
  for (int i = 0; i < 8; ++i) r[i] = 0.0f;
  return r;
}

// ---- CDNA5 async global->LDS copy (ASYNCcnt-tracked), with safe fallback ----
#if __has_builtin(__builtin_amdgcn_global_load_async_to_lds_b128)
#define HAVE_ASYNC_LDS 1
#else
#define HAVE_ASYNC_LDS 0
#endif

__device__ __forceinline__ void cp_async_b128(void* lds, const void* g) {
#if HAVE_ASYNC_LDS
  __builtin_amdgcn_global_load_async_to_lds_b128(
      (__attribute__((address_space(1))) v4i*)g,
      (__attribute__((address_space(3))) v4i*)lds, /*offset=*/0, /*cpol=*/0);
#else
  *(float4*)lds = *(const float4*)g;
#endif
}

template <int N>
__device__ __forceinline__ void wait_async() {
#if HAVE_ASYNC_LDS
  asm volatile("s_wait_asynccnt %0" ::"i"(N) : "memory");
#endif
}

// ---------------- degree / normalization ----------------
__global__ void k_deg_init(float* __restrict__ deg) {
  int i = blockIdx.x * blockDim.x + threadIdx.x;
  if (i < Nn) deg[i] = 1.0f;  // self loop
}

__global__ void k_deg_edges(const int* __restrict__ ei, float* __restrict__ deg) {
  int e = blockIdx.x * blockDim.x + threadIdx.x;
  if (e < Ee) atomicAdd(&deg[ei[Ee + e]], 1.0f);  // dst row
}

__global__ void k_dinv(const float* __restrict__ deg, float* __restrict__ dinv) {
  int i = blockIdx.x * blockDim.x + threadIdx.x;
  if (i < Nn) dinv[i] = rsqrtf(deg[i]);
}

// ---------------- generic WMMA f32 GEMM: C[M,N] = A[M,K] @ op(B) + bias ----------------
// TRANSB=0: B is [K,N] (ldb = N).  TRANSB=1: B is [N,K] row-major weights (ldb = K).
// Block: 128 threads (4 waves). Tile 128x64, K-step 16. Each wave computes 32x64
// (2 M-frags x 4 N-frags -> B frags reused). Double-buffered async LDS staging.
template <bool TRANSB, bool RELU>
__global__ __launch_bounds__(128) void k_gemm(const float* __restrict__ A,
                                              const float* __restrict__ Bm,
                                              const float* __restrict__ bias,
                                              float* __restrict__ C,
                                              int K, int ldb, int ldc) {
  __shared__ float As[2 * 2560];  // 128 rows x 16 k, stride 20 (f4-aligned, conflict-free)
  __shared__ float Bs[2 * 1280];  // !TRANSB: [16 k][64 n] stride 68; TRANSB: [64 n][16 k] stride 20

  const int t = threadIdx.x;
  const int wave = t >> 5, lane = t & 31;
  const int lh = lane & 15;
  const int koff = (lane >> 4) << 1;  // lanes 16-31 hold K+2,K+3 of A/B frags
  const int rhalf = (lane >= 16) ? 8 : 0;
  const int mb = blockIdx.x * 128, nb = blockIdx.y * 64;
  const int m0 = wave * 32 + lh;

  auto stage = [&](int kb, int buf) {
    float* Ad = As + buf * 2560;
#pragma unroll
    for (int i = 0; i < 4; ++i) {  // 512 float4
      int f = t + i * 128;
      int r = f >> 2, c4 = f & 3;
      cp_async_b128(Ad + r * 20 + c4 * 4, A + (size_t)(mb + r) * K + kb + c4 * 4);
    }
    float* Bd = Bs + buf * 1280;
    if (TRANSB) {
#pragma unroll
      for (int i = 0; i < 2; ++i) {  // 256 float4, land as [n][k]
        int f = t + i * 128;
        int n = f >> 2, k4 = f & 3;
        cp_async_b128(Bd + n * 20 + k4 * 4, Bm + (size_t)(nb + n) * ldb + kb + k4 * 4);
      }
    } else {
#pragma unroll
      for (int i = 0; i < 2; ++i) {  // 256 float4, land as [k][n]
        int f = t + i * 128;
        int k = f >> 4, n4 = f & 15;
        cp_async_b128(Bd + k * 68 + n4 * 4, Bm + (size_t)(kb + k) * ldb + nb + n4 * 4);
      }
    }
  };

  v8f acc[2][4];
#pragma unroll
  for (int mi = 0; mi < 2; ++mi)
#pragma unroll
    for (int nt = 0; nt < 4; ++nt) acc[mi][nt] = v8zero();

  const int niter = K >> 4;
  stage(0, 0);
  for (int it = 0; it < niter; ++it) {
    const int cur = it & 1;
    if (it + 1 < niter) {
      stage((it + 1) << 4, cur ^ 1);  // overlap next tile with this tile's WMMAs
      wait_async<6>();                // 6 in-order b128s just issued may stay outstanding
    } else {
      wait_async<0>();
    }
    __syncthreads();
    const float* Ab = As + cur * 2560;
    const float* Bb = Bs + cur * 1280;
#pragma unroll
    for (int kk = 0; kk < 16; kk += 4) {
      v2f a0, a1;
      a0.x = Ab[m0 * 20 + kk + koff];
      a0.y = Ab[m0 * 20 + kk + koff + 1];
      a1.x = Ab[(m0 + 16) * 20 + kk + koff];
      a1.y = Ab[(m0 + 16) * 20 + kk + koff + 1];
#pragma unroll
      for (int nt = 0; nt < 4; ++nt) {
        int n = nt * 16 + lh;
        v2f bb;
        if (TRANSB) {
          bb.x = Bb[n * 20 + kk + koff];
          bb.y = Bb[n * 20 + kk + koff + 1];
        } else {
          bb.x = Bb[(kk + koff) * 68 + n];
          bb.y = Bb[(kk + koff + 1) * 68 + n];
        }
        acc[0][nt] = wmma4(a0, bb, acc[0][nt]);
        acc[1][nt] = wmma4(a1, bb, acc[1][nt]);
      }
    }
    __syncthreads();
  }

#pragma unroll
  for (int mi = 0; mi < 2; ++mi) {
    int rowb = mb + wave * 32 + mi * 16 + rhalf;
#pragma unroll
    for (int nt = 0; nt < 4; ++nt) {
      int col = nb + nt * 16 + lh;
      float bv = bias ? bias[col] : 0.0f;
#pragma unroll
      for (int v = 0; v < 8; ++v) {
        float x = acc[mi][nt][v] + bv;
        if (RELU) x = fmaxf(x, 0.0f);
        C[(size_t)(rowb + v) * ldc + col] = x;
      }
    }
  }
}

// ---------------- GCN self-loop init + bias ----------------
__global__ void k_gcn_self(const float* __restrict__ xl, const float* __restrict__ dinv,
                           const float* __restrict__ bias, float* __restrict__ hg) {
  int t = blockIdx.x * blockDim.x + threadIdx.x;  // float4 index over B*N*H/4
  int h4 = t & 127;                                // Hh/4 = 128
  int n = (t >> 7) & (Nn - 1);
  float d = dinv[n];
  float dd = d * d;
  float4 v = ((const float4*)xl)[t];
  float4 bb = ((const float4*)bias)[h4];
  float4 r;
  r.x = bb.x + v.x * dd;
  r.y = bb.y + v.y * dd;
  r.z = bb.z + v.z * dd;
  r.w = bb.w + v.w * dd;
  ((float4*)hg)[t] = r;
}

// ---------------- GCN edge scatter (fp32 atomics) ----------------
__global__ void k_gcn_scatter(const float* __restrict__ xl, const int* __restrict__ ei,
                              const float* __restrict__ dinv, float* __restrict__ hg) {
  int t = blockIdx.x * blockDim.x + threadIdx.x;  // over Ee * (Hh/4)
  int e = t >> 7;
  int h4 = t & 127;
  int s = ei[e];
  int d = ei[Ee + e];
  float nm = dinv[s] * dinv[d];
#pragma unroll
  for (int b = 0; b < Bsz; ++b) {
    const float4 v = *(const float4*)(xl + (size_t)(b * Nn + s) * Hh + h4 * 4);
    float* dst = hg + (size_t)(b * Nn + d) * Hh + h4 * 4;
    atomicAdd(dst + 0, v.x * nm);
    atomicAdd(dst + 1, v.y * nm);
    atomicAdd(dst + 2, v.z * nm);
    atomicAdd(dst + 3, v.w * nm);
  }
}

// ---------------- flash attention: one block per (b, head, 64-query tile) ----------------
static constexpr int KT = 32;        // key tile rows
static constexpr int KVS = KT * 68;  // per-buffer K/V tile floats
__global__ __launch_bounds__(128) void k_flash(const float* __restrict__ qkv,
                                               float* __restrict__ o) {
  __shared__ float Qs[64 * 68];
  __shared__ float Ks[2 * KVS];
  __shared__ float Vs[2 * KVS];
  __shared__ float Ps[64 * 36];
  __shared__ float m_s[64], l_s[64], corr[64];

  const int t = threadIdx.x;
  const int wave = t >> 5, lane = t & 31;
  const int lh = lane & 15;
  const int koff = (lane >> 4) << 1;
  const int qbase = blockIdx.x * 64;
  const int head = blockIdx.y, b = blockIdx.z;
  const int hoff = head * DH;
  const int mrow = wave * 16 + lh;
  const int rbase = wave * 16 + ((lane >= 16) ? 8 : 0);

  auto stage_kv = [&](int kt, int buf) {
#pragma unroll
    for (int i = 0; i < 4; ++i) {  // 512 float4 per matrix
      int f = t + i * 128;
      int r = f >> 4, c4 = f & 15;
      size_t row = (size_t)(b * Nn + kt * KT + r) * (3 * Hh);
      cp_async_b128(Ks + buf * KVS + r * 68 + c4 * 4, qkv + row + Hh + hoff + c4 * 4);
      cp_async_b128(Vs + buf * KVS + r * 68 + c4 * 4, qkv + row + 2 * Hh + hoff + c4 * 4);
    }
  };

  // stage Q tile, pre-scaled by dh^-0.5 = 0.125
#pragma unroll
  for (int i = 0; i < 8; ++i) {
    int f = t + i * 128;  // 1024 float4
    int r = f >> 4, c4 = f & 15;
    float4 v = *(const float4*)(qkv + (size_t)(b * Nn + qbase + r) * (3 * Hh) + hoff + c4 * 4);
    v.x *= 0.125f; v.y *= 0.125f; v.z *= 0.125f; v.w *= 0.125f;
    *(float4*)(Qs + r * 68 + c4 * 4) = v;
  }
  if (t < 64) { m_s[t] = -3.0e38f; l_s[t] = 0.0f; }
  v8f accO[4];
#pragma unroll
  for (int i = 0; i < 4; ++i) accO[i] = v8zero();

  stage_kv(0, 0);
  const int ntile = Nn / KT;
  for (int kt = 0; kt < ntile; ++kt) {
    const int cur = kt & 1;
    if (kt + 1 < ntile) {
      stage_kv(kt + 1, cur ^ 1);  // overlap next K/V stream with this tile's math
      wait_async<8>();
    } else {
      wait_async<0>();
    }
    __syncthreads();
    const float* Kb = Ks + cur * KVS;
    const float* Vb = Vs + cur * KVS;

    // S = Q @ K^T (per-wave 16 x 32)
    v8f accS[2];
    accS[0] = v8zero();
    accS[1] = v8zero();
#pragma unroll
    for (int kk = 0; kk < DH; kk += 4) {
      v2f a;
      a.x = Qs[mrow * 68 + kk + koff];
      a.y = Qs[mrow * 68 + kk + koff + 1];
#pragma unroll
      for (int nt = 0; nt < 2; ++nt) {
        int n = nt * 16 + lh;
        v2f bb;
        bb.x = Kb[n * 68 + kk + koff];  // B[kk][n] = K[n][kk] (transposed read)
        bb.y = Kb[n * 68 + kk + koff + 1];
        accS[nt] = wmma4(a, bb, accS[nt]);
      }
    }
    // spill S (C-layout) to LDS for softmax + A-frag reload
#pragma unroll
    for (int nt = 0; nt < 2; ++nt)
#pragma unroll
      for (int v = 0; v < 8; ++v)
        Ps[(rbase + v) * 36 + nt * 16 + lh] = accS[nt][v];
    __syncthreads();

    // online softmax, one row per thread
    if (t < 64) {
      float mt = -3.0e38f;
      for (int j = 0; j < KT; ++j) mt = fmaxf(mt, Ps[t * 36 + j]);
      float mold = m_s[t];
      float mnew = fmaxf(mold, mt);
      float c = __expf(mold - mnew);
      float s = 0.0f;
      for (int j = 0; j < KT; ++j) {
        float p = __expf(Ps[t * 36 + j] - mnew);
        Ps[t * 36 + j] = p;
        s += p;
      }
      m_s[t] = mnew;
      l_s[t] = l_s[t] * c + s;
      corr[t] = c;
    }
    __syncthreads();

    // O = O*corr + P @ V
#pragma unroll
    for (int nt = 0; nt < 4; ++nt)
#pragma unroll
      for (int v = 0; v < 8; ++v) accO[nt][v] *= corr[rbase + v];
#pragma unroll
    for (int kk = 0; kk < KT; kk += 4) {
      v2f a;
      a.x = Ps[mrow * 36 + kk + koff];
      a.y = Ps[mrow * 36 + kk + koff + 1];
#pragma unroll
      for (int nt = 0; nt < 4; ++nt) {
        int n = nt * 16 + lh;
        v2f bb;
        bb.x = Vb[(kk + koff) * 68 + n];
        bb.y = Vb[(kk + koff + 1) * 68 + n];
        accO[nt] = wmma4(a, bb, accO[nt]);
      }
    }
    __syncthreads();
  }

  // normalize and write O to [B, N, H] with head offset
#pragma unroll
  for (int nt = 0; nt < 4; ++nt)
#pragma unroll
    for (int v = 0; v < 8; ++v) {
      int r = rbase + v;
      float inv = 1.0f / l_s[r];
      o[(size_t)(b * Nn + qbase + r) * Hh + hoff + nt * 16 + lh] = accO[nt][v] * inv;
    }
}

// ---------------- driver ----------------
extern "C" void kernel_launch(void* const* d_in, const int* in_sizes, int n_in,
                              void* d_out, int out_size, void* d_ws, size_t ws_size,
                              hipStream_t stream) {
  (void)in_sizes; (void)n_in; (void)out_size; (void)ws_size;
  const float* x  = (const float*)d_in[0];
  const int*   ei = (const int*)d_in[1];
  const float* Wg = (const float*)d_in[2];
  const float* bg = (const float*)d_in[3];
  const float* wi = (const float*)d_in[4];
  const float* bi = (const float*)d_in[5];
  const float* wo = (const float*)d_in[6];
  const float* bo = (const float*)d_in[7];
  const float* fw = (const float*)d_in[8];
  const float* fb = (const float*)d_in[9];
  float* out = (float*)d_out;

  float* ws   = (float*)d_ws;
  float* deg  = ws;                    // 2048
  float* dinv = ws + 2048;             // 2048
  float* xl   = ws + 4096;             // B*N*H = 4194304 (reused as h2 later)
  float* hg   = xl + 4194304;          // B*N*H
  float* qkv  = hg + 4194304;          // B*N*3H = 12582912
  float* oatt = qkv + 12582912;        // B*N*H
  float* h2   = xl;                    // alias: xl dead after scatter

  const int M = Bsz * Nn;  // 8192

  // 1) degrees + symmetric norm
  k_deg_init<<<Nn / 256, 256, 0, stream>>>(deg);
  k_deg_edges<<<Ee / 256, 256, 0, stream>>>(ei, deg);
  k_dinv<<<Nn / 256, 256, 0, stream>>>(deg, dinv);

  // 2) xl = x @ W_gcn  (bias deferred until after aggregation)
  k_gemm<false, false><<<dim3(M / 128, Hh / 64), 128, 0, stream>>>(
      x, Wg, nullptr, xl, Din, Hh, Hh);

  // 3) hg = b_gcn + xl*dinv^2 (self loops), then edge scatter-add
  k_gcn_self<<<(M * Hh / 4) / 256, 256, 0, stream>>>(xl, dinv, bg, hg);
  k_gcn_scatter<<<(Ee * (Hh / 4)) / 256, 256, 0, stream>>>(xl, ei, dinv, hg);

  // 4) qkv = hg @ in_proj_w^T + in_proj_b   -> [M, 1536]
  k_gemm<true, false><<<dim3(M / 128, (3 * Hh) / 64), 128, 0, stream>>>(
      hg, wi, bi, qkv, Hh, Hh, 3 * Hh);

  // 5) flash attention per (b, head, 64-query tile)
  k_flash<<<dim3(Nn / 64, NH, Bsz), 128, 0, stream>>>(qkv, oatt);

  // 6) h2 = relu(oatt @ out_proj_w^T + out_proj_b)
  k_gemm<true, true><<<dim3(M / 128, Hh / 64), 128, 0, stream>>>(
      oatt, wo, bo, h2, Hh, Hh, Hh);

  // 7) out = h2 @ fc_w + fc_b
  k_gemm<false, false><<<dim3(M / 128, DoutC / 64), 128, 0, stream>>>(
      h2, fw, fb, out, Hh, DoutC, DoutC);
}